// ERQHLayer_15917148799898
// MI455X (gfx1250) — compile-verified
//
#include <hip/hip_runtime.h>

// ---------------------------------------------------------------------------
// ERQH layer for MI455X (gfx1250, wave32, WMMA + async global->LDS).
// Shapes: B=4, N=8192, m=8, D=256, K=512, Q4=64.
// Memory-bound layer (~200MB HBM traffic, ~5 GFLOP) -> keep fp32 end-to-end,
// use V_WMMA_F32_16X16X4_F32 for the two quaternion-linear GEMMs, and stage
// the contiguous A panel into LDS with GLOBAL_LOAD_ASYNC_TO_LDS_B128.
// ---------------------------------------------------------------------------

typedef __attribute__((ext_vector_type(2))) float v2f;
typedef __attribute__((ext_vector_type(8))) float v8f;

static constexpr int NB      = 4;      // batch
static constexpr int NSEQ    = 8192;   // tokens per batch
static constexpr int DMODEL  = 256;
static constexpr int NCODES  = 512;
static constexpr int MSLOTS  = 8;

// ---------------------------------------------------------------------------
__global__ void zero_kernel(float* __restrict__ p, size_t n) {
  size_t i = (size_t)blockIdx.x * blockDim.x + threadIdx.x;
  size_t stride = (size_t)gridDim.x * blockDim.x;
  for (; i < n; i += stride) p[i] = 0.0f;
}

// ---------------------------------------------------------------------------
// Build transposed quaternion weight matrix: Wt[d][o] = W[o][d], where
// W = [[ r,-i,-j,-k],
//      [ i, r,-k, j],
//      [ j, k, r,-i],
//      [ k,-j, i, r]]  (64x64 blocks)
// Stored with d (GEMM K dim) as the row so B-tiles are coalesced in o.
__global__ void build_wt_kernel(const float* __restrict__ r,
                                const float* __restrict__ qi,
                                const float* __restrict__ qj,
                                const float* __restrict__ qk,
                                float* __restrict__ wt) {
  int t = blockIdx.x * blockDim.x + threadIdx.x;   // 0..65535
  int d = t >> 8;        // input feature (K dim)
  int o = t & 255;       // output feature
  int br = o >> 6, bc = d >> 6;
  int a = o & 63,  b = d & 63;
  const float* const mats[4][4] = {
    { r,  qi, qj, qk },
    { qi, r,  qk, qj },
    { qj, qk, r,  qi },
    { qk, qj, qi, r  }
  };
  const float signs[4][4] = {
    { 1.f, -1.f, -1.f, -1.f },
    { 1.f,  1.f, -1.f,  1.f },
    { 1.f,  1.f,  1.f, -1.f },
    { 1.f, -1.f,  1.f,  1.f }
  };
  wt[d * 256 + o] = signs[br][bc] * mats[br][bc][a * 64 + b];
}

// ---------------------------------------------------------------------------
// Weighted scatter of q rows into prototype accumulators (segment_sum).
// One 256-thread block per (b,n); proto table (512KB/batch) is L2-resident,
// so the f32 atomics are serviced by L2 atomic units.
__global__ void scatter_kernel(const float* __restrict__ q,
                               const int*   __restrict__ idx,
                               const float* __restrict__ aw,
                               const float* __restrict__ cmask,
                               float* __restrict__ proto_raw,
                               float* __restrict__ wsum) {
  int bn = blockIdx.x;                 // 0 .. B*N-1
  int d  = threadIdx.x;                // 0 .. 255
  int b  = bn >> 13;                   // N = 8192
  float qv = q[(size_t)bn * DMODEL + d];
  float mask = cmask[bn];
#pragma unroll
  for (int s = 0; s < MSLOTS; ++s) {
    int   code = idx[bn * MSLOTS + s];
    float w    = aw[bn * MSLOTS + s] * mask;
    atomicAdd(&proto_raw[((size_t)b * NCODES + code) * DMODEL + d], w * qv);
    if (d == 0) atomicAdd(&wsum[b * NCODES + code], w);
  }
}

// ---------------------------------------------------------------------------
// Fp32 WMMA GEMM:  Y[R x 256] = X[R x 256] @ Wt[256 x 256] + bias
// One block per 16-row M-tile. The 16x256 A panel is contiguous in memory ->
// staged once into LDS via async global->LDS b128 copies (ASYNCcnt-tracked),
// then 8 waves each compute two 16x16 N-tiles (64 v_wmma each) reading A
// from LDS and B from the L2-resident Wt.
// Optional per-row scale 1/(wsum[row]+1e-6) (prototype normalization).
// A layout (ISA 16x4 f32): lanes 0-15 -> M=lane, K = k,k+1 in vgpr0/1;
//                          lanes 16-31 -> same M,  K = k+2,k+3.
// B layout mirrored: vgpr0 = row K (lanes 0-15) / K+2 (lanes 16-31).
// C/D: 8 VGPRs, vgpr r: M = r (lanes 0-15) / r+8 (lanes 16-31), N = lane%16.
__global__ void gemm_wmma_256(const float* __restrict__ X,
                              const float* __restrict__ Wt,
                              const float* __restrict__ bias,
                              const float* __restrict__ wsum,   // nullable
                              float* __restrict__ Y,
                              float* __restrict__ Y2) {          // nullable
  __shared__ float sA[16 * 256];       // 16 KB A panel
  int m0 = blockIdx.x * 16;

  // --- async-stage the contiguous 16x256 A panel into LDS ---
  // 256 threads x 4 iters x b128 (4 floats) = 4096 floats = 16 KB.
  // Generic LDS pointers carry the LDS byte offset in addr[31:0] (ISA 10.2),
  // which is what the async instruction's LDS-address VGPR wants.
  {
    const float* src = X + (size_t)m0 * 256;
    unsigned ldsbase = (unsigned)(uintptr_t)sA;
#pragma unroll
    for (int it = 0; it < 4; ++it) {
      int e = threadIdx.x * 4 + it * 1024;       // float index in panel
      unsigned laddr = ldsbase + (unsigned)(e * 4);
      const float* g = src + e;
      asm volatile("global_load_async_to_lds_b128 %0, %1, off"
                   :: "v"(laddr), "v"(g) : "memory");
    }
    asm volatile("s_wait_asynccnt 0x0" ::: "memory");
  }
  __syncthreads();

  int wave = threadIdx.x >> 5;        // 0..7 -> two N-tiles each
  int lane = threadIdx.x & 31;
  int half = lane >> 4;               // 0: K pair (k,k+1), 1: (k+2,k+3)
  int l16  = lane & 15;
  int n0a  = wave * 32;
  int n0b  = n0a + 16;

  float scale = 1.0f;
  if (wsum) scale = 1.0f / (wsum[m0 + l16] + 1e-6f);
  const float* arow = sA + l16 * 256;

  v8f acc0 = {}, acc1 = {};
#pragma unroll 4
  for (int k = 0; k < 256; k += 4) {
    int ak = k + half * 2;
    v2f a, b0, b1;
    a.x  = arow[ak]     * scale;      // ds_load from LDS panel
    a.y  = arow[ak + 1] * scale;
    b0.x = Wt[(size_t)ak       * 256 + n0a + l16];
    b0.y = Wt[(size_t)(ak + 1) * 256 + n0a + l16];
    b1.x = Wt[(size_t)ak       * 256 + n0b + l16];
    b1.y = Wt[(size_t)(ak + 1) * 256 + n0b + l16];
    acc0 = __builtin_amdgcn_wmma_f32_16x16x4_f32(
        false, a, false, b0, (short)0, acc0, false, false);
    acc1 = __builtin_amdgcn_wmma_f32_16x16x4_f32(
        false, a, false, b1, (short)0, acc1, false, false);
  }

#pragma unroll
  for (int rr = 0; rr < 8; ++rr) {
    int row = m0 + rr + half * 8;
    float va = acc0[rr] + bias[n0a + l16];
    float vb = acc1[rr] + bias[n0b + l16];
    Y[(size_t)row * 256 + n0a + l16] = va;
    Y[(size_t)row * 256 + n0b + l16] = vb;
    if (Y2) {
      Y2[(size_t)row * 256 + n0a + l16] = va;
      Y2[(size_t)row * 256 + n0b + l16] = vb;
    }
  }
}

// ---------------------------------------------------------------------------
// Routing: msg[b,n,:] = sum_s assign_w[b,n,s] * hamilton(q[b,n,:], proto[b,idx])
// One wave per (b,n); each lane owns 2 quaternion components (lane, lane+32).
// Loads within each 64-float component block are fully coalesced; the gathered
// prototypes (512KB/batch) hit L2.
__global__ void route_kernel(const float* __restrict__ q,
                             const int*   __restrict__ idx,
                             const float* __restrict__ aw,
                             const float* __restrict__ proto,
                             float* __restrict__ msg) {
  int bn   = (blockIdx.x * blockDim.x + threadIdx.x) >> 5;
  int lane = threadIdx.x & 31;
  int b    = bn >> 13;
  const float* qrow = q + (size_t)bn * DMODEL;

  float pr0 = qrow[lane],        pr1 = qrow[lane + 32];
  float pi0 = qrow[64 + lane],   pi1 = qrow[96 + lane];
  float pj0 = qrow[128 + lane],  pj1 = qrow[160 + lane];
  float pk0 = qrow[192 + lane],  pk1 = qrow[224 + lane];

  float mr0 = 0, mi0 = 0, mj0 = 0, mk0 = 0;
  float mr1 = 0, mi1 = 0, mj1 = 0, mk1 = 0;

#pragma unroll
  for (int s = 0; s < MSLOTS; ++s) {
    int   code = idx[bn * MSLOTS + s];
    float w    = aw[bn * MSLOTS + s];
    const float* pp = proto + ((size_t)b * NCODES + code) * DMODEL;
    float qr0 = pp[lane],       qr1 = pp[lane + 32];
    float qi0 = pp[64 + lane],  qi1 = pp[96 + lane];
    float qj0 = pp[128 + lane], qj1 = pp[160 + lane];
    float qk0 = pp[192 + lane], qk1 = pp[224 + lane];

    mr0 += w * (pr0 * qr0 - pi0 * qi0 - pj0 * qj0 - pk0 * qk0);
    mi0 += w * (pr0 * qi0 + pi0 * qr0 + pj0 * qk0 - pk0 * qj0);
    mj0 += w * (pr0 * qj0 - pi0 * qk0 + pj0 * qr0 + pk0 * qi0);
    mk0 += w * (pr0 * qk0 + pi0 * qj0 - pj0 * qi0 + pk0 * qr0);

    mr1 += w * (pr1 * qr1 - pi1 * qi1 - pj1 * qj1 - pk1 * qk1);
    mi1 += w * (pr1 * qi1 + pi1 * qr1 + pj1 * qk1 - pk1 * qj1);
    mj1 += w * (pr1 * qj1 - pi1 * qk1 + pj1 * qr1 + pk1 * qi1);
    mk1 += w * (pr1 * qk1 + pi1 * qj1 - pj1 * qi1 + pk1 * qr1);
  }

  float* mrow = msg + (size_t)bn * DMODEL;
  mrow[lane]        = mr0;  mrow[lane + 32]   = mr1;
  mrow[64 + lane]   = mi0;  mrow[96 + lane]   = mi1;
  mrow[128 + lane]  = mj0;  mrow[160 + lane]  = mj1;
  mrow[192 + lane]  = mk0;  mrow[224 + lane]  = mk1;
}

// ---------------------------------------------------------------------------
// Residual + quaternion LayerNorm over each of the 4 groups of 64.
// One wave per (b,n); 2 elements per lane per group; wave32 butterfly reduce.
__global__ void add_ln_kernel(const float* __restrict__ q,
                              const float* __restrict__ msg_lin,
                              const float* __restrict__ gamma,
                              const float* __restrict__ beta,
                              float* __restrict__ out) {
  int bn   = (blockIdx.x * blockDim.x + threadIdx.x) >> 5;
  int lane = threadIdx.x & 31;
  const float* qrow = q       + (size_t)bn * DMODEL;
  const float* mrow = msg_lin + (size_t)bn * DMODEL;
  float* orow       = out     + (size_t)bn * DMODEL;

#pragma unroll
  for (int g = 0; g < 4; ++g) {
    int c0 = g * 64 + lane, c1 = c0 + 32;
    float x0 = qrow[c0] + mrow[c0];
    float x1 = qrow[c1] + mrow[c1];
    float s  = x0 + x1;
    float sq = x0 * x0 + x1 * x1;
#pragma unroll
    for (int off = 16; off > 0; off >>= 1) {
      s  += __shfl_xor(s, off, 32);
      sq += __shfl_xor(sq, off, 32);
    }
    float mean = s * (1.0f / 64.0f);
    float var  = sq * (1.0f / 64.0f) - mean * mean;
    float inv  = rsqrtf(var + 1e-5f);
    orow[c0] = (x0 - mean) * inv * gamma[g * 64 + lane]      + beta[g * 64 + lane];
    orow[c1] = (x1 - mean) * inv * gamma[g * 64 + lane + 32] + beta[g * 64 + lane + 32];
  }
}

// ---------------------------------------------------------------------------
extern "C" void kernel_launch(void* const* d_in, const int* in_sizes, int n_in,
                              void* d_out, int out_size, void* d_ws, size_t ws_size,
                              hipStream_t stream) {
  const float* q     = (const float*)d_in[0];
  const int*   idx   = (const int*)  d_in[1];
  const float* aw    = (const float*)d_in[2];
  const float* cmask = (const float*)d_in[3];
  const float* pr    = (const float*)d_in[4];
  const float* pi    = (const float*)d_in[5];
  const float* pj    = (const float*)d_in[6];
  const float* pk    = (const float*)d_in[7];
  const float* pb    = (const float*)d_in[8];
  const float* ur    = (const float*)d_in[9];
  const float* ui    = (const float*)d_in[10];
  const float* uj    = (const float*)d_in[11];
  const float* uk    = (const float*)d_in[12];
  const float* ub    = (const float*)d_in[13];
  const float* gamma = (const float*)d_in[14];
  const float* beta  = (const float*)d_in[15];
  float* out = (float*)d_out;

  // Workspace layout (floats)
  const size_t SZ_PROTO = (size_t)NB * NCODES * DMODEL;   // 524288
  const size_t SZ_WSUM  = (size_t)NB * NCODES;            // 2048
  const size_t SZ_WT    = 256 * 256;                      // 65536
  const size_t SZ_MSG   = (size_t)NB * NSEQ * DMODEL;     // 8388608

  float* ws        = (float*)d_ws;
  float* proto_raw = ws;
  float* wsum      = proto_raw + SZ_PROTO;
  float* wt_proto  = wsum + SZ_WSUM;
  float* wt_upd    = wt_proto + SZ_WT;
  float* proto_fin = wt_upd + SZ_WT;
  float* msg       = proto_fin + SZ_PROTO;
  float* msg_lin   = msg + SZ_MSG;

  float* qnew_out  = out;                    // (B,N,D)
  float* proto_out = out + SZ_MSG;           // (B,K,D)

  // 1) zero scatter accumulators (ws is poisoned by harness)
  zero_kernel<<<512, 256, 0, stream>>>(proto_raw, SZ_PROTO + SZ_WSUM);

  // 2) build transposed quaternion weight matrices
  build_wt_kernel<<<256, 256, 0, stream>>>(pr, pi, pj, pk, wt_proto);
  build_wt_kernel<<<256, 256, 0, stream>>>(ur, ui, uj, uk, wt_upd);

  // 3) weighted segment-sum scatter
  scatter_kernel<<<NB * NSEQ, 256, 0, stream>>>(q, idx, aw, cmask,
                                                proto_raw, wsum);

  // 4) normalize + quat_linear(proto) via WMMA: 2048 rows -> 128 M-tiles
  gemm_wmma_256<<<(NB * NCODES) / 16, 256, 0, stream>>>(
      proto_raw, wt_proto, pb, wsum, proto_fin, proto_out);

  // 5) gather + hamilton + weighted sum
  route_kernel<<<(NB * NSEQ) / 8, 256, 0, stream>>>(q, idx, aw, proto_fin, msg);

  // 6) quat_linear(msg) via WMMA: 32768 rows -> 2048 M-tiles
  gemm_wmma_256<<<(NB * NSEQ) / 16, 256, 0, stream>>>(
      msg, wt_upd, ub, nullptr, msg_lin, nullptr);

  // 7) residual + quaternion LayerNorm
  add_ln_kernel<<<(NB * NSEQ) / 8, 256, 0, stream>>>(q, msg_lin, gamma, beta,
                                                     qnew_out);
}